// filter_1_output_1_64905545777705
// MI455X (gfx1250) — compile-verified
//
#include <hip/hip_runtime.h>

typedef __attribute__((ext_vector_type(2))) float v2f;
typedef __attribute__((ext_vector_type(8))) float v8f;

#define N_ATOMS 2048
#define RBF_DIM 32

__device__ __forceinline__ float softplus_f(float x) {
  // numerically stable softplus: max(x,0) + log1p(exp(-|x|))
  return fmaxf(x, 0.0f) + log1pf(expf(-fabsf(x)));
}

__global__ void zero_kernel(float* __restrict__ out, int n) {
  int i = blockIdx.x * blockDim.x + threadIdx.x;
  if (i < n) out[i] = 0.0f;
}

// One block = one 'a' row x 128 b's. 8 waves/block, each wave owns 16 (a,b) pairs.
// Per wave: fp32 WMMA chain (16x16x4, K=32, two N-halves) -> softplus -> dot(W2)
// -> shuffle-reduce -> mask/unit/cross -> LDS-atomic reduce -> global atomicAdd.
//
// W1 is staged in LDS in a WMMA-B-fragment-swizzled layout so each lane's
// B fragment (K,K+1 pair for a given k-step/half) is one contiguous b64 load:
//   sB[((ki*2 + h)*32 + lane)*2 + e] = W1[4*ki + 2*(lane>>4) + e][16*h + (lane&15)]
// (bijection over all 1024 W1 elements; ds_load_b64, lane stride 8B => no bank
// conflicts).
__global__ __launch_bounds__(256) void f1_fused_kernel(
    const float* __restrict__ layer_input,  // [N,1,3]
    const float* __restrict__ rbf,          // [N,N,32]
    const float* __restrict__ rij,          // [N,N,3]
    const float* __restrict__ W1,           // [32,32] (r,h) row-major
    const float* __restrict__ b1,           // [32]
    const float* __restrict__ W2,           // [32,1]
    const float* __restrict__ b2,           // [1]
    float* __restrict__ out)                // [N,1,3]
{
  __shared__ __align__(16) float sB[RBF_DIM * RBF_DIM];  // swizzled W1 (4 KB)
  __shared__ float sb1[RBF_DIM];
  __shared__ float sW2[RBF_DIM];
  __shared__ float acc[3];

  const int tid = threadIdx.x;
  // Swizzled preload of W1 into B-fragment order.
#pragma unroll
  for (int t = tid; t < RBF_DIM * RBF_DIM; t += 256) {
    const int e      = t & 1;
    const int lane_t = (t >> 1) & 31;
    const int h      = (t >> 6) & 1;
    const int ki     = t >> 7;             // 0..7
    const int r      = 4 * ki + 2 * (lane_t >> 4) + e;
    const int c      = 16 * h + (lane_t & 15);
    sB[t] = W1[r * RBF_DIM + c];
  }
  if (tid < RBF_DIM) { sb1[tid] = b1[tid]; sW2[tid] = W2[tid]; }
  if (tid < 3) acc[tid] = 0.0f;
  __syncthreads();

  const int a      = blockIdx.x >> 4;     // 2048 a-rows
  const int bchunk = blockIdx.x & 15;     // 16 chunks of 128 b's
  const int wave   = tid >> 5;            // 8 waves
  const int lane   = tid & 31;
  const int b0     = bchunk * 128 + wave * 16;

  const int n    = lane & 15;             // A: row M; B: column N; C: N index
  const int g    = lane >> 4;             // lane half-group
  const int koff = g * 2;                 // A/B K sub-offset per ISA layout

  // A-matrix row for this lane: rbf[a, b0+n, :]
  const float* arow = rbf + ((size_t)a * N_ATOMS + (size_t)(b0 + n)) * RBF_DIM;
  const float* bfrag = &sB[lane * 2];     // + ki*64 + h*32 floats

  v8f c0 = {};  // hidden dims 0..15
  v8f c1 = {};  // hidden dims 16..31
#pragma unroll
  for (int ki = 0; ki < 8; ++ki) {
    const int k = 4 * ki;
    // A 16x4 fragment: lane n, VGPR0/1 = K=k+koff, k+koff+1 (contiguous pair)
    v2f av = __builtin_nontemporal_load((const v2f*)(arow + k + koff));
    // B 4x16 fragments: single contiguous b64 each from swizzled LDS
    v2f bv0 = *(const v2f*)(bfrag + ki * 64);
    v2f bv1 = *(const v2f*)(bfrag + ki * 64 + 32);
    c0 = __builtin_amdgcn_wmma_f32_16x16x4_f32(false, av, false, bv0,
                                               (short)0, c0, false, false);
    c1 = __builtin_amdgcn_wmma_f32_16x16x4_f32(false, av, false, bv1,
                                               (short)0, c1, false, false);
  }

  // C layout: reg r, lane-group g -> pair M = r + 8*g, hidden dim = n + 16*half
  const float bb0 = sb1[n],  bb1 = sb1[n + 16];
  const float w20 = sW2[n],  w21 = sW2[n + 16];
  const float b2s = b2[0];
  float part[8];
#pragma unroll
  for (int r = 0; r < 8; ++r) {
    float h0 = softplus_f(c0[r] + bb0);
    float h1 = softplus_f(c1[r] + bb1);
    part[r] = h0 * w20 + h1 * w21;
  }
  // reduce over the 16 lanes of each half-group (butterfly)
#pragma unroll
  for (int m = 1; m <= 8; m <<= 1) {
#pragma unroll
    for (int r = 0; r < 8; ++r) part[r] += __shfl_xor(part[r], m, 32);
  }

  // lanes 0..7 handle pairs 0..7 (g=0); lanes 16..23 handle pairs 8..15 (g=1)
  if (n < 8) {
    float rad = part[0];
#pragma unroll
    for (int j = 1; j < 8; ++j) rad = (n == j) ? part[j] : rad;
    rad += b2s;

    const int b = b0 + g * 8 + n;
    const float* rp = rij + ((size_t)a * N_ATOMS + (size_t)b) * 3;
    float x = rp[0], y = rp[1], z = rp[2];
    float ss = x * x + y * y + z * z;
    // mask: ||rij|| < 1e-8  <=>  ss < 1e-16 ; unit: rij / sqrt(max(ss,1e-8))
    float s = (ss < 1e-16f) ? 0.0f : rad / sqrtf(fmaxf(ss, 1e-8f));
    const float* lp = layer_input + (size_t)b * 3;
    float lx = lp[0], ly = lp[1], lz = lp[2];
    // eps_ijk F1_j L_k = (F1 x L)_i
    atomicAdd(&acc[0], s * (y * lz - z * ly));
    atomicAdd(&acc[1], s * (z * lx - x * lz));
    atomicAdd(&acc[2], s * (x * ly - y * lx));
  }
  __syncthreads();
  if (tid < 3) atomicAdd(&out[a * 3 + tid], acc[tid]);
}

extern "C" void kernel_launch(void* const* d_in, const int* in_sizes, int n_in,
                              void* d_out, int out_size, void* d_ws, size_t ws_size,
                              hipStream_t stream) {
  const float* layer_input = (const float*)d_in[0];
  const float* rbf         = (const float*)d_in[1];
  const float* rij         = (const float*)d_in[2];
  const float* W1          = (const float*)d_in[3];
  const float* b1          = (const float*)d_in[4];
  const float* W2          = (const float*)d_in[5];
  const float* b2          = (const float*)d_in[6];
  float* out = (float*)d_out;

  zero_kernel<<<(out_size + 255) / 256, 256, 0, stream>>>(out, out_size);

  dim3 grid(N_ATOMS * 16);  // 2048 a-rows x 16 b-chunks
  f1_fused_kernel<<<grid, 256, 0, stream>>>(layer_input, rbf, rij,
                                            W1, b1, W2, b2, out);
}